// GaussRenderer_67826123538551
// MI455X (gfx1250) — compile-verified
//
#include <hip/hip_runtime.h>
#include <math.h>

#define IMG 128
#define TILEW 32
#define NPOS 2048
#define NNEG 8
#define CHUNK 16
#define NCHUNK (NPOS / CHUNK)

typedef __attribute__((ext_vector_type(2))) float v2f;
typedef __attribute__((ext_vector_type(8))) float v8f;

// ---------------------------------------------------------------------------
// Kernel 1: bitonic argsort of depths (N=2048) in LDS, one 1024-thread block.
// ---------------------------------------------------------------------------
__global__ __launch_bounds__(1024) void sort_kernel(const float* __restrict__ depths,
                                                    int* __restrict__ sIdx,
                                                    float* __restrict__ sDep) {
    __shared__ float key[NPOS];
    __shared__ int   idx[NPOS];
    const int t = threadIdx.x;
    for (int i = t; i < NPOS; i += 1024) { key[i] = depths[i]; idx[i] = i; }
    __syncthreads();
    for (int k = 2; k <= NPOS; k <<= 1) {
        for (int j = k >> 1; j > 0; j >>= 1) {
            for (int i = t; i < NPOS; i += 1024) {
                int p = i ^ j;
                if (p > i) {
                    bool up = ((i & k) == 0);
                    float ki = key[i], kp = key[p];
                    if ((ki > kp) == up) {
                        key[i] = kp; key[p] = ki;
                        int ti = idx[i]; idx[i] = idx[p]; idx[p] = ti;
                    }
                }
            }
            __syncthreads();
        }
    }
    for (int i = t; i < NPOS; i += 1024) { sIdx[i] = idx[i]; sDep[i] = key[i]; }
}

// ---------------------------------------------------------------------------
// Kernel 2: per-Gaussian prep in sorted order.
// conic = inv(cov2d); quadratic-form coefficients with -0.5 folded in:
//   exponent(x,y) = k0*x^2 + k1*y^2 + k2*x*y + (mean-dependent terms computed
//   per-tile in the render kernel to keep f32 cancellation small)
// ---------------------------------------------------------------------------
__global__ __launch_bounds__(256) void prep_pos_kernel(
    const float* __restrict__ means, const float* __restrict__ cov,
    const float* __restrict__ color, const float* __restrict__ op,
    const int* __restrict__ sIdx,
    float* __restrict__ k0, float* __restrict__ k1, float* __restrict__ k2,
    float* __restrict__ mx, float* __restrict__ my,
    float* __restrict__ cr, float* __restrict__ cg, float* __restrict__ cb,
    float* __restrict__ sop,
    float* __restrict__ rx0, float* __restrict__ ry0,
    float* __restrict__ rx1, float* __restrict__ ry1) {
    int j = blockIdx.x * blockDim.x + threadIdx.x;
    if (j >= NPOS) return;
    int i = sIdx[j];
    float c00 = cov[i * 4 + 0], c01 = cov[i * 4 + 1];
    float c10 = cov[i * 4 + 2], c11 = cov[i * 4 + 3];
    float det = c00 * c11 - c01 * c10;
    float inv = 1.0f / det;
    float A  = c11 * inv;            // conic00
    float Bv = c00 * inv;            // conic11
    float Cc = -(c01 + c10) * inv;   // conic01 + conic10
    k0[j] = -0.5f * A;
    k1[j] = -0.5f * Bv;
    k2[j] = -0.5f * Cc;
    float mxv = means[i * 2 + 0], myv = means[i * 2 + 1];
    mx[j] = mxv; my[j] = myv;
    cr[j] = color[i * 3 + 0]; cg[j] = color[i * 3 + 1]; cb[j] = color[i * 3 + 2];
    sop[j] = op[i];
    // 3-sigma radius: lam = mid + sqrt(max(mid^2 - det, 0.1))
    float mid = 0.5f * (c00 + c11);
    float s = sqrtf(fmaxf(mid * mid - det, 0.1f));
    float r = 3.0f * ceilf(sqrtf(mid + s));
    rx0[j] = fminf(fmaxf(mxv - r, 0.0f), (float)(IMG - 1));
    ry0[j] = fminf(fmaxf(myv - r, 0.0f), (float)(IMG - 1));
    rx1[j] = fminf(fmaxf(mxv + r, 0.0f), (float)(IMG - 1));
    ry1[j] = fminf(fmaxf(myv + r, 0.0f), (float)(IMG - 1));
}

// ---------------------------------------------------------------------------
// Kernel 3: negative-Gaussian prep (8 of them): full 3x3 inverse + rects.
// negp[m*14]: {mx, my, nc00, nc11, nc01+nc10, nc02+nc20, nc12+nc21, nc22,
//              depth, opacity, rx0, ry0, rx1, ry1}
// ---------------------------------------------------------------------------
__global__ void prep_neg_kernel(const float* __restrict__ nm, const float* __restrict__ ncov,
                                const float* __restrict__ nop, const float* __restrict__ nd,
                                float* __restrict__ negp) {
    int m = threadIdx.x;
    if (m >= NNEG) return;
    const float* c = ncov + m * 9;
    float a = c[0], b = c[1], cc = c[2];
    float d = c[3], e = c[4], f  = c[5];
    float g = c[6], h = c[7], ii = c[8];
    float det = a * (e * ii - f * h) - b * (d * ii - f * g) + cc * (d * h - e * g);
    float inv = 1.0f / det;
    float i00 = (e * ii - f * h) * inv, i01 = (cc * h - b * ii) * inv, i02 = (b * f - cc * e) * inv;
    float i10 = (f * g - d * ii) * inv, i11 = (a * ii - cc * g) * inv, i12 = (cc * d - a * f) * inv;
    float i20 = (d * h - e * g) * inv,  i21 = (b * g - a * h) * inv,   i22 = (a * e - b * d) * inv;
    float mx = nm[m * 2 + 0], my = nm[m * 2 + 1];
    // radius from top-left 2x2 of the covariance (not its inverse)
    float det2 = a * e - b * d;
    float mid = 0.5f * (a + e);
    float s = sqrtf(fmaxf(mid * mid - det2, 0.1f));
    float r = 3.0f * ceilf(sqrtf(mid + s));
    float* q = negp + m * 14;
    q[0] = mx; q[1] = my;
    q[2] = i00; q[3] = i11; q[4] = i01 + i10; q[5] = i02 + i20; q[6] = i12 + i21; q[7] = i22;
    q[8] = nd[m]; q[9] = nop[m];
    q[10] = fminf(fmaxf(mx - r, 0.0f), (float)(IMG - 1));
    q[11] = fminf(fmaxf(my - r, 0.0f), (float)(IMG - 1));
    q[12] = fminf(fmaxf(mx + r, 0.0f), (float)(IMG - 1));
    q[13] = fminf(fmaxf(my + r, 0.0f), (float)(IMG - 1));
}

// ---------------------------------------------------------------------------
// Kernel 4: render. One block = one 16x16 pixel sub-block of a 32x32 tile.
// Exponent blocks computed with V_WMMA_F32_16X16X4_F32 (K=8 via two chained
// WMMAs): A = pixel features [x2, y2, xy, x | y, 1, 0, 0] (tile-centered),
// B = per-Gaussian coefficients. Then alpha/exp on the trans pipe (co-executes
// with XDL WMMA), front-to-back compositing one thread per pixel.
// ---------------------------------------------------------------------------
__global__ __launch_bounds__(256) void render_kernel(
    const float* __restrict__ sDep,
    const float* __restrict__ k0a, const float* __restrict__ k1a, const float* __restrict__ k2a,
    const float* __restrict__ mxa, const float* __restrict__ mya,
    const float* __restrict__ cra, const float* __restrict__ cga, const float* __restrict__ cba,
    const float* __restrict__ opa,
    const float* __restrict__ rx0a, const float* __restrict__ ry0a,
    const float* __restrict__ rx1a, const float* __restrict__ ry1a,
    const float* __restrict__ negp,
    float* __restrict__ out) {
    __shared__ float sAlpha[256 * 17];   // [pixel][n], padded stride 17 (gcd(17,64)=1)
    __shared__ float sK0[CHUNK], sK1[CHUNK], sK2[CHUNK], sK3[CHUNK], sK4[CHUNK], sK5[CHUNK];
    __shared__ float sCr[CHUNK], sCg[CHUNK], sCb[CHUNK], sD[CHUNK], sOp[CHUNK], sM[CHUNK];

    const int tid  = threadIdx.x;
    const int blk  = blockIdx.x;
    const int tile = blk >> 2, sb = blk & 3;
    const int th = (tile >> 2) * TILEW, tw = (tile & 3) * TILEW;
    const int sy = (sb >> 1) * 16, sx = (sb & 1) * 16;
    const float ox = (float)tw + 15.5f, oy = (float)th + 15.5f;   // tile center

    const int pyl = tid >> 4, pxl = tid & 15;
    const int py = th + sy + pyl, px = tw + sx + pxl;
    const float fx = (float)px, fy = (float)py;

    // --- per-pixel negative-Gaussian precompute (quad/lin in pixel only) ---
    float nQ[NNEG], nL[NNEG], nC[NNEG], nD[NNEG], nO[NNEG];
    bool anyNeg = false;
#pragma unroll
    for (int m = 0; m < NNEG; ++m) {
        const float* q = negp + m * 14;
        float nmx = q[0], nmy = q[1];
        float a = q[2], b = q[3], cxy = q[4], cx = q[5], cy = q[6], czz = q[7];
        bool ov = (fminf(q[12], (float)(tw + TILEW - 1)) > fmaxf(q[10], (float)tw)) &&
                  (fminf(q[13], (float)(th + TILEW - 1)) > fmaxf(q[11], (float)th));
        float dnx = fx - nmx, dny = fy - nmy;
        nQ[m] = dnx * dnx * a + dny * dny * b + dnx * dny * cxy;
        nL[m] = dnx * cx + dny * cy;
        nC[m] = czz;
        nD[m] = q[8];
        nO[m] = ov ? q[9] : 0.0f;   // where(nmask, ., 0): zero opacity kills term
        anyNeg = anyNeg || ov;      // uniform across the block (tile-level)
    }

    // --- per-wave constant WMMA A operands (pixel features, tile-centered) ---
    // A 16x4 f32 layout: lanes 0-15 hold M=lane with K=0(v0),1(v1);
    //                    lanes 16-31 hold M=lane-16 with K=2(v0),3(v1).
    const int wave = tid >> 5, lane = tid & 31;
    const int hv = lane >> 4, Ml = lane & 15;
    const float axl = (float)(tw + sx + Ml) - ox;              // pixel x of row M
    const float ay0 = (float)(th + sy + wave) - oy;            // group g0 = wave
    const float ay1 = (float)(th + sy + wave + 8) - oy;        // group g1 = wave+8
    v2f A00, A10, A01, A11;
    if (hv == 0) {
        A00.x = axl * axl; A00.y = ay0 * ay0;   // K0=x^2, K1=y^2
        A10.x = ay0;       A10.y = 1.0f;        // K4=y,   K5=1
        A01.x = axl * axl; A01.y = ay1 * ay1;
        A11.x = ay1;       A11.y = 1.0f;
    } else {
        A00.x = axl * ay0; A00.y = axl;         // K2=xy,  K3=x
        A10.x = 0.0f;      A10.y = 0.0f;        // K6=K7=0
        A01.x = axl * ay1; A01.y = axl;
        A11.x = 0.0f;      A11.y = 0.0f;
    }
    const int nlane = lane & 15;
    const int pBase0 = wave * 16;          // pixel index p = g*16 + M
    const int pBase1 = (wave + 8) * 16;

    float T = 1.0f, cr = 0.0f, cg = 0.0f, cb = 0.0f, dp = 0.0f, ac = 0.0f;

    for (int ch = 0; ch < NCHUNK; ++ch) {
        __syncthreads();
        if (tid < CHUNK) {
            int i = ch * CHUNK + tid;
            float K0 = k0a[i], K1 = k1a[i], K2 = k2a[i];
            float mxl = mxa[i] - ox, myl = mya[i] - oy;
            float Am = -2.0f * K0, Bm = -2.0f * K1, Cm = -2.0f * K2;
            sK0[tid] = K0; sK1[tid] = K1; sK2[tid] = K2;
            sK3[tid] = Am * mxl + 0.5f * Cm * myl;
            sK4[tid] = Bm * myl + 0.5f * Cm * mxl;
            sK5[tid] = -0.5f * (Am * mxl * mxl + Bm * myl * myl + Cm * mxl * myl);
            sCr[tid] = cra[i]; sCg[tid] = cga[i]; sCb[tid] = cba[i];
            sD[tid] = sDep[i]; sOp[tid] = opa[i];
            bool ov = (fminf(rx1a[i], (float)(tw + TILEW - 1)) > fmaxf(rx0a[i], (float)tw)) &&
                      (fminf(ry1a[i], (float)(th + TILEW - 1)) > fmaxf(ry0a[i], (float)th));
            sM[tid] = ov ? 1.0f : 0.0f;
            if (ch + 1 < NCHUNK) {  // global_prefetch_b8 for next chunk
                int ip = i + CHUNK;
                __builtin_prefetch(&k0a[ip], 0, 1);
                __builtin_prefetch(&mxa[ip], 0, 1);
                __builtin_prefetch(&cra[ip], 0, 1);
            }
        }
        __syncthreads();

        // --- WMMA exponent blocks: D(16 pix x 16 gauss) = A(16x8) * B(8x16) ---
        // B 4x16 layout: lanes 0-15 hold N=lane with K=0(v0),1(v1);
        //                lanes 16-31 hold N=lane-16 with K=2(v0),3(v1).
        v2f B0, B1;
        if (hv == 0) { B0.x = sK0[nlane]; B0.y = sK1[nlane]; B1.x = sK4[nlane]; B1.y = sK5[nlane]; }
        else         { B0.x = sK2[nlane]; B0.y = sK3[nlane]; B1.x = 0.0f;       B1.y = 0.0f; }

        v8f e0 = {0.f, 0.f, 0.f, 0.f, 0.f, 0.f, 0.f, 0.f};
        e0 = __builtin_amdgcn_wmma_f32_16x16x4_f32(false, A00, false, B0, (short)0, e0, false, false);
        e0 = __builtin_amdgcn_wmma_f32_16x16x4_f32(false, A10, false, B1, (short)0, e0, false, false);
        v8f e1 = {0.f, 0.f, 0.f, 0.f, 0.f, 0.f, 0.f, 0.f};
        e1 = __builtin_amdgcn_wmma_f32_16x16x4_f32(false, A01, false, B0, (short)0, e1, false, false);
        e1 = __builtin_amdgcn_wmma_f32_16x16x4_f32(false, A11, false, B1, (short)0, e1, false, false);

        // D layout: VGPR v, lane half h -> row M = v + 8h, col N = lane&15
        float opn = sOp[nlane];
        float mkn = sM[nlane];
#pragma unroll
        for (int v = 0; v < 8; ++v) {
            int M = v + 8 * hv;
            float a0 = (mkn != 0.0f) ? fminf(__expf(e0[v]) * opn, 0.99f) : 0.0f;
            float a1 = (mkn != 0.0f) ? fminf(__expf(e1[v]) * opn, 0.99f) : 0.0f;
            sAlpha[(pBase0 + M) * 17 + nlane] = a0;
            sAlpha[(pBase1 + M) * 17 + nlane] = a1;
        }
        __syncthreads();

        // --- front-to-back compositing: one thread per pixel ---
        for (int i = 0; i < CHUNK; ++i) {
            float a = sAlpha[tid * 17 + i];
            if (anyNeg && sM[i] != 0.0f) {
                float dgi = sD[i];
                float na = 0.0f;
#pragma unroll
                for (int m = 0; m < NNEG; ++m) {
                    if (nO[m] != 0.0f) {
                        float dz = dgi - nD[m];
                        float ee = nQ[m] + (nL[m] + nC[m] * dz) * dz;
                        na += fminf(__expf(-0.5f * ee) * nO[m], 0.99f);
                    }
                }
                a = fmaxf(a - na, 0.01f);
            }
            float wgt = a * T;
            cr += wgt * sCr[i]; cg += wgt * sCg[i]; cb += wgt * sCb[i];
            dp += wgt * sD[i];  ac += wgt;
            T *= (1.0f - a);
        }
    }

    const int o = py * IMG + px;
    const float bk = 1.0f - ac;          // white background
    out[o * 3 + 0] = cr + bk;
    out[o * 3 + 1] = cg + bk;
    out[o * 3 + 2] = cb + bk;
    out[IMG * IMG * 3 + o] = dp;
    out[IMG * IMG * 4 + o] = ac;
}

// ---------------------------------------------------------------------------
extern "C" void kernel_launch(void* const* d_in, const int* in_sizes, int n_in,
                              void* d_out, int out_size, void* d_ws, size_t ws_size,
                              hipStream_t stream) {
    const float* means   = (const float*)d_in[0];
    const float* cov2d   = (const float*)d_in[1];
    const float* color   = (const float*)d_in[2];
    const float* opacity = (const float*)d_in[3];
    const float* depths  = (const float*)d_in[4];
    const float* nmeans  = (const float*)d_in[5];
    const float* ncov    = (const float*)d_in[6];
    const float* nop     = (const float*)d_in[7];
    const float* ndep    = (const float*)d_in[8];
    float* out = (float*)d_out;

    // workspace layout: 2048*15 floats + 112 floats (~123 KB)
    float* w   = (float*)d_ws;
    int*   sIdx = (int*)w;                 // [2048] (int)
    float* sDep = w + NPOS;                // sorted depths
    float* k0  = w + 2 * NPOS;
    float* k1  = k0 + NPOS;
    float* k2  = k1 + NPOS;
    float* mx  = k2 + NPOS;
    float* my  = mx + NPOS;
    float* cr  = my + NPOS;
    float* cg  = cr + NPOS;
    float* cb  = cg + NPOS;
    float* op  = cb + NPOS;
    float* rx0 = op + NPOS;
    float* ry0 = rx0 + NPOS;
    float* rx1 = ry0 + NPOS;
    float* ry1 = rx1 + NPOS;
    float* negp = ry1 + NPOS;              // [8*14]

    sort_kernel<<<1, 1024, 0, stream>>>(depths, sIdx, sDep);
    prep_pos_kernel<<<(NPOS + 255) / 256, 256, 0, stream>>>(
        means, cov2d, color, opacity, sIdx,
        k0, k1, k2, mx, my, cr, cg, cb, op, rx0, ry0, rx1, ry1);
    prep_neg_kernel<<<1, 32, 0, stream>>>(nmeans, ncov, nop, ndep, negp);
    render_kernel<<<64, 256, 0, stream>>>(
        sDep, k0, k1, k2, mx, my, cr, cg, cb, op, rx0, ry0, rx1, ry1, negp, out);
}